// GNN_32873679684030
// MI455X (gfx1250) — compile-verified
//
#include <hip/hip_runtime.h>

#define THREADS 256

typedef __attribute__((ext_vector_type(2))) float v2f;
typedef __attribute__((ext_vector_type(8))) float v8f;

// Guaranteed hardware L2-side float atomic (non-returning, device scope).
__device__ __forceinline__ void atomic_fadd_dev(float* p, float v) {
  asm volatile("global_atomic_add_f32 %0, %1, off scope:SCOPE_DEV"
               : : "v"(p), "v"(v) : "memory");
}

// Butterfly add step via ds_swizzle immediate pattern (xor within 32 lanes).
// IMM must be a compile-time constant for the builtin.
template <int IMM>
__device__ __forceinline__ float swz_add(float t) {
  return t + __int_as_float(__builtin_amdgcn_ds_swizzle(__float_as_int(t), IMM));
}

__global__ void zero_b32(unsigned* __restrict__ p, long n) {
  long i = (long)blockIdx.x * blockDim.x + threadIdx.x;
  if (i < n) p[i] = 0u;
}

// deg[col[e]] += 1 (u32). Self-loop +1 added in prep_kernel.
__global__ void degree_kernel(const int* __restrict__ col, unsigned* __restrict__ deg, int E) {
  int e = blockIdx.x * blockDim.x + threadIdx.x;
  if (e < E) {
    int c = __builtin_nontemporal_load(col + e);   // NT: don't thrash L2
    atomicAdd(&deg[c], 1u);
  }
}

// In-place u32 degree -> f32 dinv = rsqrt(deg+1); also ys4[n] = {x[n]*dinv, 0}
__global__ void prep_kernel(const float* __restrict__ x, float* __restrict__ dinv,
                            float4* __restrict__ ys4, int N) {
  int n = blockIdx.x * blockDim.x + threadIdx.x;
  if (n >= N) return;
  unsigned d = ((const unsigned*)dinv)[n] + 1u;    // +1 = self loop
  float di = rsqrtf((float)d);
  dinv[n] = di;
  float4 y;
  y.x = x[(size_t)n * 3 + 0] * di;
  y.y = x[(size_t)n * 3 + 1] * di;
  y.z = x[(size_t)n * 3 + 2] * di;
  y.w = 0.0f;
  ys4[n] = y;
}

// Layer-1 edge scatter in 3-feature space: acc3[col][k] += ys[row][k]*dinv[col]
__global__ void edge_l1(const int* __restrict__ row, const int* __restrict__ col,
                        const float* __restrict__ dinv, const float4* __restrict__ ys4,
                        float* __restrict__ acc3, int E) {
  int e = blockIdx.x * blockDim.x + threadIdx.x;
  if (e >= E) return;
  int r = __builtin_nontemporal_load(row + e);
  int c = __builtin_nontemporal_load(col + e);
  float w = dinv[c];
  float4 y = ys4[r];                 // one aligned b128 gather (L2-resident)
  float* a = acc3 + (size_t)c * 4;
  atomic_fadd_dev(a + 0, y.x * w);
  atomic_fadd_dev(a + 1, y.y * w);
  atomic_fadd_dev(a + 2, y.z * w);
}

// Fused mid kernel (full 16-node tiles, branchless):
//   Atot[n][k] = acc3[n][k] + ys[n][k]*dinv[n]  (self loop; k=3 is 0-padded)
//   D = Atot @ W1  via V_WMMA_F32_16X16X4_F32
//   zs[n] = (relu(D[n]+b1) . W2) * dinv[n]      (ds_swizzle butterfly reduce)
__global__ __launch_bounds__(THREADS) void wmma_mid(
    const float* __restrict__ acc3, const float* __restrict__ ys,
    const float* __restrict__ dinv, const float* __restrict__ W1,
    const float* __restrict__ b1, const float* __restrict__ W2,
    float* __restrict__ zs, int ftiles) {
  int wave = (int)((blockIdx.x * blockDim.x + threadIdx.x) >> 5);
  if (wave >= ftiles) return;                 // wave-uniform
  int lane = threadIdx.x & 31;
  int m    = lane & 15;
  int half = lane >> 4;
  int kb   = half * 2;                        // K base 0 or 2
  int node = wave * 16 + m;

  float di = dinv[node];
  const float2 av = *(const float2*)(acc3 + (size_t)node * 4 + kb);  // 8B aligned
  const float2 yv = *(const float2*)(ys   + (size_t)node * 4 + kb);
  v2f a;
  a.x = fmaf(yv.x, di, av.x);                 // K=kb
  a.y = fmaf(yv.y, di, av.y);                 // K=kb+1 (kb=2 -> both arrays 0)

  // B: W1 [3][8] row-major, pad N>=8 and K=3 with cndmask (no branches)
  int mc = m & 7;
  float w1a = W1[kb * 8 + mc];                             // rows 0/2, always valid
  float w1b = W1[((kb + 1 < 3) ? (kb + 1) : kb) * 8 + mc]; // clamped index
  v2f b;
  b.x = (m < 8) ? w1a : 0.0f;
  b.y = (m < 8 && kb + 1 < 3) ? w1b : 0.0f;

  v8f c = {};
  c = __builtin_amdgcn_wmma_f32_16x16x4_f32(
      false, a, false, b, (short)0, c, false, false);

  float b1v = b1[mc];
  float w2v = W2[mc];
  float zres = 0.0f;
#pragma unroll
  for (int r = 0; r < 8; ++r) {
    // lane m holds D[M = r+8*half][n = m]; cols n>=8 contribute 0
    float t = (m < 8) ? fmaxf(c[r] + b1v, 0.0f) * w2v : 0.0f;
    t = swz_add<0x041F>(t);                   // xor 1
    t = swz_add<0x081F>(t);                   // xor 2
    t = swz_add<0x101F>(t);                   // xor 4
    t = swz_add<0x201F>(t);                   // xor 8 -> sum over 16-lane half
    zres = (m == r) ? t : zres;               // cndmask capture, no branch
  }
  if (m < 8) {                                // lanes 0-7 / 16-23: coalesced store
    int sn = wave * 16 + m + half * 8;
    zs[sn] = zres * dinv[sn];
  }
}

// Scalar tail for N % 16 nodes (not launched when N is a multiple of 16).
__global__ void tail_mid(const float* __restrict__ acc3, const float* __restrict__ ys,
                         const float* __restrict__ dinv, const float* __restrict__ W1,
                         const float* __restrict__ b1, const float* __restrict__ W2,
                         float* __restrict__ zs, int start, int N) {
  int n = start + blockIdx.x * blockDim.x + threadIdx.x;
  if (n >= N) return;
  float di = dinv[n];
  float a0 = fmaf(ys[(size_t)n * 4 + 0], di, acc3[(size_t)n * 4 + 0]);
  float a1 = fmaf(ys[(size_t)n * 4 + 1], di, acc3[(size_t)n * 4 + 1]);
  float a2 = fmaf(ys[(size_t)n * 4 + 2], di, acc3[(size_t)n * 4 + 2]);
  float z = 0.0f;
#pragma unroll
  for (int j = 0; j < 8; ++j) {
    float h = a0 * W1[j] + a1 * W1[8 + j] + a2 * W1[16 + j] + b1[j];
    z = fmaf(fmaxf(h, 0.0f), W2[j], z);
  }
  zs[n] = z * di;
}

// Layer-2 edge scatter: out[col] += zs[row] * dinv[col]
__global__ void edge_l2(const int* __restrict__ row, const int* __restrict__ col,
                        const float* __restrict__ dinv, const float* __restrict__ zs,
                        float* __restrict__ out, int E) {
  int e = blockIdx.x * blockDim.x + threadIdx.x;
  if (e >= E) return;
  int r = __builtin_nontemporal_load(row + e);
  int c = __builtin_nontemporal_load(col + e);
  atomic_fadd_dev(&out[c], zs[r] * dinv[c]);
}

// out += self_loop + b2
__global__ void finalize_kernel(const float* __restrict__ zs, const float* __restrict__ dinv,
                                const float* __restrict__ b2, float* __restrict__ out, int N) {
  int n = blockIdx.x * blockDim.x + threadIdx.x;
  if (n < N) out[n] += zs[n] * dinv[n] + b2[0];
}

extern "C" void kernel_launch(void* const* d_in, const int* in_sizes, int n_in,
                              void* d_out, int out_size, void* d_ws, size_t ws_size,
                              hipStream_t stream) {
  const float* x  = (const float*)d_in[0];
  const int*   ei = (const int*)d_in[1];
  const float* W1 = (const float*)d_in[2];
  const float* b1 = (const float*)d_in[3];
  const float* W2 = (const float*)d_in[4];
  const float* b2 = (const float*)d_in[5];

  const int N = in_sizes[0] / 3;
  const int E = in_sizes[1] / 2;
  const int* row = ei;       // source
  const int* col = ei + E;   // target

  float* out = (float*)d_out;

  // Workspace: dinv[N] (u32 deg first), ys[4N], acc3[4N], zs[N] = 10N*4 B (40 MB)
  float* dinv = (float*)d_ws;
  float* ys   = dinv + N;
  float* acc3 = ys + (size_t)4 * N;
  float* zs   = acc3 + (size_t)4 * N;

  const int gN  = (N + THREADS - 1) / THREADS;
  const int gE  = (E + THREADS - 1) / THREADS;
  const long n4 = (long)4 * N;
  const int g4N = (int)((n4 + THREADS - 1) / THREADS);

  const int ftiles = N / 16;                 // full WMMA tiles
  const int tailStart = ftiles * 16;
  const int tailCnt = N - tailStart;
  const int gW = (ftiles * 32 + THREADS - 1) / THREADS;

  zero_b32<<<gN,  THREADS, 0, stream>>>((unsigned*)dinv, (long)N);
  zero_b32<<<g4N, THREADS, 0, stream>>>((unsigned*)acc3, n4);
  zero_b32<<<gN,  THREADS, 0, stream>>>((unsigned*)out,  (long)N);

  degree_kernel<<<gE, THREADS, 0, stream>>>(col, (unsigned*)dinv, E);
  prep_kernel  <<<gN, THREADS, 0, stream>>>(x, dinv, (float4*)ys, N);

  edge_l1<<<gE, THREADS, 0, stream>>>(row, col, dinv, (const float4*)ys, acc3, E);

  if (ftiles > 0)
    wmma_mid<<<gW, THREADS, 0, stream>>>(acc3, ys, dinv, W1, b1, W2, zs, ftiles);
  if (tailCnt > 0)
    tail_mid<<<1, 32, 0, stream>>>(acc3, ys, dinv, W1, b1, W2, zs, tailStart, N);

  edge_l2<<<gE, THREADS, 0, stream>>>(row, col, dinv, zs, out, E);
  finalize_kernel<<<gN, THREADS, 0, stream>>>(zs, dinv, b2, out, N);
}